// GCNEncoder_72395968741626
// MI455X (gfx1250) — compile-verified
//
#include <hip/hip_runtime.h>
#include <cstdint>

typedef __attribute__((ext_vector_type(2))) float v2f;
typedef __attribute__((ext_vector_type(8))) float v8f;

// ---------------------------------------------------------------------------
// Degree / normalization kernels
// ---------------------------------------------------------------------------
__global__ void k_deg_init(float* __restrict__ deg, int M) {
  int i = blockIdx.x * blockDim.x + threadIdx.x;
  if (i < M) deg[i] = 1.0f;  // self-loop contributes 1 to every node's degree
}

__global__ void k_deg_accum(const long long* __restrict__ dst,
                            float* __restrict__ deg, int E) {
  int e = blockIdx.x * blockDim.x + threadIdx.x;
  if (e < E) {
    int d = (int)dst[e];
    unsafeAtomicAdd(&deg[d], 1.0f);  // native global_atomic_add_f32
  }
}

__global__ void k_deg_rsqrt(float* __restrict__ deg, int M) {
  int i = blockIdx.x * blockDim.x + threadIdx.x;
  if (i < M) {
    float d = deg[i];
    deg[i] = (d > 0.0f) ? rsqrtf(d) : 0.0f;  // becomes dis[i]
  }
}

// ---------------------------------------------------------------------------
// fp32 WMMA GEMM: C[M,N] = A[M,K] @ B[K,N]   (optionally ReLU on A reads)
// One wave32 computes a 16x64 output tile: 4 x v_wmma_f32_16x16x4_f32 accums.
// K, N are compile-time constants so all B accesses reduce to one 32-bit
// voffset plus constant immediate offsets, and k*N strength-reduces fully.
// A-row is clamped (not masked): garbage rows only feed accumulator rows
// whose stores are guarded, so no per-iteration EXEC manipulation is needed.
// ---------------------------------------------------------------------------
template <int K, int N, bool RELU_A>
__global__ __launch_bounds__(256) void k_gemm_wmma_f32(
    const float* __restrict__ A, const float* __restrict__ B,
    float* __restrict__ C, int M) {
  const int lane = threadIdx.x & 31;
  const int wid = (blockIdx.x * blockDim.x + threadIdx.x) >> 5;
  const int mtiles = (M + 15) >> 4;
  constexpr int ngroups = N >> 6;
  if (wid >= mtiles * ngroups) return;  // wave-uniform exit: EXEC stays all-1s

  const int mt = wid % mtiles;
  const int ng = wid / mtiles;

  const int lrow = lane & 15;  // M (for A) / N (for B,C) within tile
  const int hi = lane >> 4;    // half-wave select
  const int khi = hi << 1;     // K offset 0 or 2 for A/B fragments

  const int row = (mt << 4) + lrow;
  const int crow = row < M ? row : (M - 1);  // clamp: safe, stores are guarded
  const int ncol0 = (ng << 6) + lrow;

  // 32-bit element offsets -> saddr + 32-bit voffset addressing
  const int aoff0 = crow * K + khi;
  const int boff0 = khi * N + ncol0;

  v8f acc0 = {}, acc1 = {}, acc2 = {}, acc3 = {};

#pragma unroll 8
  for (int k = 0; k < K; k += 4) {
    // A fragment: lane holds A[row][k+khi], A[row][k+khi+1]  (ISA 16x4 layout)
    v2f a = *(const v2f*)(A + (aoff0 + k));
    if (RELU_A) {
      a.x = fmaxf(a.x, 0.0f);
      a.y = fmaxf(a.y, 0.0f);
    }
    // B fragments: vgpr0 = B[k+khi][n], vgpr1 = B[k+khi+1][n]  (4x16 layout)
    const int b0 = boff0 + k * N;  // k*N: constant-step strength reduction
    v2f f0, f1, f2, f3;
    f0.x = B[b0];
    f0.y = B[b0 + N];
    f1.x = B[b0 + 16];
    f1.y = B[b0 + N + 16];
    f2.x = B[b0 + 32];
    f2.y = B[b0 + N + 32];
    f3.x = B[b0 + 48];
    f3.y = B[b0 + N + 48];

    acc0 = __builtin_amdgcn_wmma_f32_16x16x4_f32(false, a, false, f0, (short)0,
                                                 acc0, false, false);
    acc1 = __builtin_amdgcn_wmma_f32_16x16x4_f32(false, a, false, f1, (short)0,
                                                 acc1, false, false);
    acc2 = __builtin_amdgcn_wmma_f32_16x16x4_f32(false, a, false, f2, (short)0,
                                                 acc2, false, false);
    acc3 = __builtin_amdgcn_wmma_f32_16x16x4_f32(false, a, false, f3, (short)0,
                                                 acc3, false, false);
  }

  // Store: C/D layout -> vgpr v, lanes<16: M=v, lanes>=16: M=v+8, N=lane&15
  const int rbase = (mt << 4) + (hi << 3);
#pragma unroll
  for (int v = 0; v < 8; ++v) {
    int r = rbase + v;
    if (r < M) {
      float* Crow = C + (r * N + ncol0);
      Crow[0] = acc0[v];
      Crow[16] = acc1[v];
      Crow[32] = acc2[v];
      Crow[48] = acc3[v];
    }
  }
}

// ---------------------------------------------------------------------------
// agg[i][c] = bias[c] + dis[i]^2 * h[i][c]   (bias + folded self-loop term)
// F is a power of two: fshift = log2(F).
// ---------------------------------------------------------------------------
__global__ void k_init_selfbias(const float* __restrict__ h,
                                const float* __restrict__ dis,
                                const float* __restrict__ bias,
                                float* __restrict__ out, int total,
                                int fshift) {
  int idx = blockIdx.x * blockDim.x + threadIdx.x;
  if (idx < total) {
    int i = idx >> fshift;
    int c = idx & ((1 << fshift) - 1);
    float d = dis[i];
    out[idx] = bias[c] + d * d * h[idx];
  }
}

// ---------------------------------------------------------------------------
// Edge scatter: out[dst] += dis[src]*dis[dst] * h[src]; one wave32 per edge.
// h and out both L2-resident (<=51 MB each) -> atomics stay on-chip.
// ---------------------------------------------------------------------------
__global__ __launch_bounds__(256) void k_edge_scatter(
    const long long* __restrict__ src, const long long* __restrict__ dst,
    const float* __restrict__ dis, const float* __restrict__ h,
    float* __restrict__ out, int E, int F) {
  const int lane = threadIdx.x & 31;
  const int wid = (blockIdx.x * blockDim.x + threadIdx.x) >> 5;
  if (wid >= E) return;
  const int s = (int)src[wid];  // wave-uniform -> scalar loads
  const int d = (int)dst[wid];
  const float nrm = dis[s] * dis[d];
  const int hbase = s * F;
  const int obase = d * F;
  for (int c = lane; c < F; c += 32) {
    unsafeAtomicAdd(out + (obase + c), nrm * h[hbase + c]);
  }
}

// ---------------------------------------------------------------------------
// Final: y = log_softmax(relu(h)) over 64 channels; one wave32 per node.
// ---------------------------------------------------------------------------
__global__ __launch_bounds__(256) void k_relu_logsoftmax64(
    float* __restrict__ out, int M) {
  const int lane = threadIdx.x & 31;
  const int node = (blockIdx.x * blockDim.x + threadIdx.x) >> 5;
  if (node >= M) return;
  float* row = out + node * 64;
  float z0 = fmaxf(row[lane], 0.0f);
  float z1 = fmaxf(row[lane + 32], 0.0f);
  float m = fmaxf(z0, z1);
#pragma unroll
  for (int off = 16; off > 0; off >>= 1) m = fmaxf(m, __shfl_xor(m, off, 32));
  float s = expf(z0 - m) + expf(z1 - m);
#pragma unroll
  for (int off = 16; off > 0; off >>= 1) s += __shfl_xor(s, off, 32);
  float lse = m + logf(s);
  row[lane] = z0 - lse;
  row[lane + 32] = z1 - lse;
}

// ---------------------------------------------------------------------------
// Host launcher
// ---------------------------------------------------------------------------
extern "C" void kernel_launch(void* const* d_in, const int* in_sizes, int n_in,
                              void* d_out, int out_size, void* d_ws,
                              size_t ws_size, hipStream_t stream) {
  const float* x = (const float*)d_in[0];
  const long long* ei = (const long long*)d_in[1];  // int64 [2, E]
  const float* W1 = (const float*)d_in[2];
  const float* b1 = (const float*)d_in[3];
  const float* W2 = (const float*)d_in[4];
  const float* b2 = (const float*)d_in[5];
  const float* W3 = (const float*)d_in[6];
  const float* b3 = (const float*)d_in[7];
  float* out = (float*)d_out;

  constexpr int K1 = 512, N1 = 256, N2 = 128, N3 = 64;
  const int M = in_sizes[0] / K1;  // 50000 nodes
  const int E = in_sizes[1] / 2;   // 800000 edges

  const long long* srcI = ei;      // edge_index[0]
  const long long* dstI = ei + E;  // edge_index[1]

  // Workspace: dis[M] | bufH (max M*256 f32) | bufA (max M*256 f32)
  char* ws = (char*)d_ws;
  float* dis = (float*)ws;
  size_t off = (((size_t)M * 4) + 255) & ~(size_t)255;
  float* bufH = (float*)(ws + off);
  float* bufA = (float*)(ws + off + (size_t)M * N1 * sizeof(float));

  const int TB = 256;
  // --- degree -> dis ---
  k_deg_init<<<(M + TB - 1) / TB, TB, 0, stream>>>(dis, M);
  k_deg_accum<<<(E + TB - 1) / TB, TB, 0, stream>>>(dstI, dis, E);
  k_deg_rsqrt<<<(M + TB - 1) / TB, TB, 0, stream>>>(dis, M);

  const int mtiles = (M + 15) / 16;
  auto gemm_blocks = [&](int N) { return (mtiles * (N / 64) + 7) / 8; };
  const int eblocks = (E + 7) / 8;  // one wave per edge, 8 waves per block

  // --- Layer 1: h1 = x @ W1 ; agg1 = b1 + self + scatter ---
  k_gemm_wmma_f32<K1, N1, false>
      <<<gemm_blocks(N1), TB, 0, stream>>>(x, W1, bufH, M);
  k_init_selfbias<<<(M * N1 + TB - 1) / TB, TB, 0, stream>>>(bufH, dis, b1,
                                                             bufA, M * N1, 8);
  k_edge_scatter<<<eblocks, TB, 0, stream>>>(srcI, dstI, dis, bufH, bufA, E, N1);

  // --- Layer 2: h2 = relu(agg1) @ W2 (relu fused into A reads) ---
  k_gemm_wmma_f32<N1, N2, true>
      <<<gemm_blocks(N2), TB, 0, stream>>>(bufA, W2, bufH, M);
  k_init_selfbias<<<(M * N2 + TB - 1) / TB, TB, 0, stream>>>(bufH, dis, b2,
                                                             bufA, M * N2, 7);
  k_edge_scatter<<<eblocks, TB, 0, stream>>>(srcI, dstI, dis, bufH, bufA, E, N2);

  // --- Layer 3: h3 = relu(agg2) @ W3 ; aggregate straight into d_out ---
  k_gemm_wmma_f32<N2, N3, true>
      <<<gemm_blocks(N3), TB, 0, stream>>>(bufA, W3, bufH, M);
  k_init_selfbias<<<(M * N3 + TB - 1) / TB, TB, 0, stream>>>(bufH, dis, b3,
                                                             out, M * N3, 6);
  k_edge_scatter<<<eblocks, TB, 0, stream>>>(srcI, dstI, dis, bufH, out, E, N3);

  // --- final relu + log_softmax in place on d_out ---
  k_relu_logsoftmax64<<<(M + 7) / 8, TB, 0, stream>>>(out, M);
}